// GNNLayer_9036611191398
// MI455X (gfx1250) — compile-verified
//
#include <hip/hip_runtime.h>

#define LRELU_SLOPE 0.2f

typedef float v2f __attribute__((ext_vector_type(2)));
typedef float v8f __attribute__((ext_vector_type(8)));

// ---------------------------------------------------------------------------
// One wave computes a 16x64 tile of  M = X[16x64] @ W[64x64]  with
// V_WMMA_F32_16X16X4_F32 (4 N-tiles, K accumulated in steps of 4).
// Fused epilogue: PL[row] = M[row,:] . aL   and  PR[row] = M[row,:] . aR
// using the C/D VGPR layout (vgpr r -> row r+8*half, lane -> column) and
// intra-half shfl_xor reductions (wave32).
// ---------------------------------------------------------------------------
__global__ __launch_bounds__(32) void gemm64_wmma_proj(
    const float* __restrict__ X, const float* __restrict__ W,
    const float* __restrict__ aL, const float* __restrict__ aR,
    float* __restrict__ M, float* __restrict__ PL, float* __restrict__ PR,
    int nrows)
{
    const int row0 = blockIdx.x << 4;
    const int lane = threadIdx.x & 31;
    const int half = lane >> 4;     // 0: K=0,1 / rows 0-7 of D ; 1: K=2,3 / rows 8-15
    const int l16  = lane & 15;

    v8f acc0 = {}, acc1 = {}, acc2 = {}, acc3 = {};

    int ar = row0 + l16;
    if (ar >= nrows) ar = nrows - 1;          // clamp (tail tile)
    const float* xrow = X + (size_t)ar * 64;

    for (int k0 = 0; k0 < 64; k0 += 4) {
        // A fragment: lane half 0 -> K=k0,k0+1 ; half 1 -> K=k0+2,k0+3 (rows = l16)
        v2f a;
        a.x = xrow[k0 + 2 * half + 0];
        a.y = xrow[k0 + 2 * half + 1];
        // B fragments: row K striped across lanes; two K rows per lane (v2f)
        const float* wk = W + (size_t)(k0 + 2 * half) * 64 + l16;
        v2f b0; b0.x = wk[0];  b0.y = wk[64 + 0];
        v2f b1; b1.x = wk[16]; b1.y = wk[64 + 16];
        v2f b2; b2.x = wk[32]; b2.y = wk[64 + 32];
        v2f b3; b3.x = wk[48]; b3.y = wk[64 + 48];

        acc0 = __builtin_amdgcn_wmma_f32_16x16x4_f32(false, a, false, b0, (short)0, acc0, false, false);
        acc1 = __builtin_amdgcn_wmma_f32_16x16x4_f32(false, a, false, b1, (short)0, acc1, false, false);
        acc2 = __builtin_amdgcn_wmma_f32_16x16x4_f32(false, a, false, b2, (short)0, acc2, false, false);
        acc3 = __builtin_amdgcn_wmma_f32_16x16x4_f32(false, a, false, b3, (short)0, acc3, false, false);
    }

    // Store D: vgpr r, lanes0-15 -> (row0+r, N=l16), lanes16-31 -> (row0+r+8, N=l16)
    #pragma unroll
    for (int r = 0; r < 8; ++r) {
        const int row = row0 + r + 8 * half;
        if (row < nrows) {
            float* orow = M + (size_t)row * 64 + l16;
            orow[0]  = acc0[r];
            orow[16] = acc1[r];
            orow[32] = acc2[r];
            orow[48] = acc3[r];
        }
    }

    // Fused projections (GEMV epilogue)
    if (aL || aR) {
        const float aL0 = aL ? aL[l16]      : 0.f;
        const float aL1 = aL ? aL[l16 + 16] : 0.f;
        const float aL2 = aL ? aL[l16 + 32] : 0.f;
        const float aL3 = aL ? aL[l16 + 48] : 0.f;
        const float aR0 = aR ? aR[l16]      : 0.f;
        const float aR1 = aR ? aR[l16 + 16] : 0.f;
        const float aR2 = aR ? aR[l16 + 32] : 0.f;
        const float aR3 = aR ? aR[l16 + 48] : 0.f;
        #pragma unroll
        for (int r = 0; r < 8; ++r) {
            float sL = acc0[r] * aL0 + acc1[r] * aL1 + acc2[r] * aL2 + acc3[r] * aL3;
            float sR = acc0[r] * aR0 + acc1[r] * aR1 + acc2[r] * aR2 + acc3[r] * aR3;
            // reduce over the 16 lanes of this half (xor masks < 16 stay in-half)
            #pragma unroll
            for (int m = 1; m < 16; m <<= 1) {
                sL += __shfl_xor(sL, m, 32);
                sR += __shfl_xor(sR, m, 32);
            }
            const int row = row0 + r + 8 * half;
            if (l16 == 0 && row < nrows) {
                if (aL && PL) PL[row] = sL;
                if (aR && PR) PR[row] = sR;
            }
        }
    }
}

// ---------------------------------------------------------------------------
// HBS edge scoring: e = leaky_relu(pl[i] + pr[j]); row-sum into RS[i].
// ---------------------------------------------------------------------------
__global__ __launch_bounds__(256) void edge_score_hbs(
    const int* __restrict__ I, const int* __restrict__ J,
    const float* __restrict__ PL, const float* __restrict__ PR,
    float* __restrict__ E, float* __restrict__ RS, int nE)
{
    int k = blockIdx.x * blockDim.x + threadIdx.x;
    if (k >= nE) return;
    const int i = I[k], j = J[k];
    const float s = PL[i] + PR[j];
    const float e = (s > 0.f) ? s : LRELU_SLOPE * s;
    E[k] = e;
    atomicAdd(&RS[i], e);
}

// ---------------------------------------------------------------------------
// HBNS edge scoring: e = leaky_relu(ps[s] + pt[t]); row-sums for both
// normalizations (per-target and per-source).
// ---------------------------------------------------------------------------
__global__ __launch_bounds__(256) void edge_score_hbns(
    const int* __restrict__ T, const int* __restrict__ S,
    const float* __restrict__ PT, const float* __restrict__ PS,
    float* __restrict__ E, float* __restrict__ RST, float* __restrict__ RSS, int nE)
{
    int k = blockIdx.x * blockDim.x + threadIdx.x;
    if (k >= nE) return;
    const int t = T[k], s = S[k];
    const float v = PS[s] + PT[t];
    const float e = (v > 0.f) ? v : LRELU_SLOPE * v;
    E[k] = e;
    atomicAdd(&RST[t], e);
    atomicAdd(&RSS[s], e);
}

// ---------------------------------------------------------------------------
// Scatter: OUT[TGT[k], :] += (E[k]/RS[TGT[k]]) * M[SRC[k], :]
// One wave32 per edge, 2 columns per lane.
// ---------------------------------------------------------------------------
__global__ __launch_bounds__(256) void edge_scatter(
    const int* __restrict__ TGT, const int* __restrict__ SRC,
    const float* __restrict__ E, const float* __restrict__ RS,
    const float* __restrict__ M, float* __restrict__ OUT, int nE)
{
    const int wid  = (blockIdx.x * blockDim.x + threadIdx.x) >> 5;
    const int lane = threadIdx.x & 31;
    if (wid >= nE) return;
    const int i = TGT[wid], j = SRC[wid];
    float rs = RS[i];
    rs = (rs == 0.f) ? 1.f : rs;
    const float att = E[wid] / rs;
    const float* src = M   + (size_t)j * 64;
    float*       dst = OUT + (size_t)i * 64;
    atomicAdd(&dst[lane],      att * src[lane]);
    atomicAdd(&dst[lane + 32], att * src[lane + 32]);
}

// ---------------------------------------------------------------------------
extern "C" void kernel_launch(void* const* d_in, const int* in_sizes, int n_in,
                              void* d_out, int out_size, void* d_ws, size_t ws_size,
                              hipStream_t stream)
{
    (void)n_in; (void)out_size; (void)ws_size;

    const float* x0   = (const float*)d_in[0];
    const float* x1   = (const float*)d_in[1];
    const float* x2   = (const float*)d_in[2];
    const float* x3   = (const float*)d_in[3];
    const float* x4   = (const float*)d_in[4];
    const int*   adj0 = (const int*)d_in[5];   // [2,E0] flat: rows i then cols j
    const int*   adj1 = (const int*)d_in[6];   // [2,E1]
    const int*   inct = (const int*)d_in[7];   // [E01] targets (0-cells)
    const int*   incs = (const int*)d_in[8];   // [E01] sources (1-cells)
    const float* W0   = (const float*)d_in[9];
    const float* a0   = (const float*)d_in[10];
    const float* W1   = (const float*)d_in[11];
    const float* a1   = (const float*)d_in[12];
    const float* Ws   = (const float*)d_in[13];
    const float* Wt   = (const float*)d_in[14];
    const float* ans  = (const float*)d_in[15];

    const int N0  = in_sizes[0] / 64;
    const int N1  = in_sizes[1] / 64;
    const int E0  = in_sizes[5] / 2;
    const int E1  = in_sizes[6] / 2;
    const int E01 = in_sizes[7];

    // --------------------- workspace layout (floats) ----------------------
    float* ws = (float*)d_ws;
    size_t off = 0;
    float* m0  = ws + off; off += (size_t)N0 * 64;   // x0 @ W0
    float* m1  = ws + off; off += (size_t)N1 * 64;   // x1 @ W1
    float* sm  = ws + off; off += (size_t)N1 * 64;   // x1 @ Ws
    float* tm  = ws + off; off += (size_t)N0 * 64;   // x0 @ Wt
    float* p0l = ws + off; off += (size_t)N0;
    float* p0r = ws + off; off += (size_t)N0;
    float* p1l = ws + off; off += (size_t)N1;
    float* p1r = ws + off; off += (size_t)N1;
    float* ps  = ws + off; off += (size_t)N1;        // sm @ ans[:64]
    float* pt  = ws + off; off += (size_t)N0;        // tm @ ans[64:]
    float* e0  = ws + off; off += (size_t)E0;
    float* e1  = ws + off; off += (size_t)E1;
    float* e01 = ws + off; off += (size_t)E01;
    float* rs0 = ws + off; off += (size_t)N0;        // |-- contiguous zero block
    float* rs1 = ws + off; off += (size_t)N1;        // |
    float* rst = ws + off; off += (size_t)N0;        // |
    float* rss = ws + off; off += (size_t)N1;        // |

    // --------------------------- outputs -----------------------------------
    float* out0 = (float*)d_out;                 // x_0_level1 [N0,64]
    float* out1 = out0 + (size_t)N0 * 64;        // x_1_level1 [N1,64]
    float* out2 = out1 + (size_t)N1 * 64;
    float* out3 = out2 + (size_t)in_sizes[2];
    float* out4 = out3 + (size_t)in_sizes[3];

    // zero accumulators (capture-safe)
    hipMemsetAsync(out0, 0, (size_t)(N0 + N1) * 64 * sizeof(float), stream);
    hipMemsetAsync(rs0, 0, (size_t)(2 * N0 + 2 * N1) * sizeof(float), stream);

    // pass-through outputs
    hipMemcpyAsync(out2, x2, (size_t)in_sizes[2] * sizeof(float), hipMemcpyDeviceToDevice, stream);
    hipMemcpyAsync(out3, x3, (size_t)in_sizes[3] * sizeof(float), hipMemcpyDeviceToDevice, stream);
    hipMemcpyAsync(out4, x4, (size_t)in_sizes[4] * sizeof(float), hipMemcpyDeviceToDevice, stream);

    // ----------------------- dense phase (WMMA) ----------------------------
    const int g0 = (N0 + 15) / 16, g1 = (N1 + 15) / 16;
    gemm64_wmma_proj<<<g0, 32, 0, stream>>>(x0, W0, a0,      a0 + 64, m0, p0l, p0r, N0);
    gemm64_wmma_proj<<<g1, 32, 0, stream>>>(x1, W1, a1,      a1 + 64, m1, p1l, p1r, N1);
    gemm64_wmma_proj<<<g1, 32, 0, stream>>>(x1, Ws, ans,     nullptr, sm, ps,  nullptr, N1);
    gemm64_wmma_proj<<<g0, 32, 0, stream>>>(x0, Wt, ans + 64, nullptr, tm, pt,  nullptr, N0);

    // ----------------------- edge scoring + row sums -----------------------
    edge_score_hbs <<<(E0  + 255) / 256, 256, 0, stream>>>(adj0, adj0 + E0, p0l, p0r, e0,  rs0, E0);
    edge_score_hbs <<<(E1  + 255) / 256, 256, 0, stream>>>(adj1, adj1 + E1, p1l, p1r, e1,  rs1, E1);
    edge_score_hbns<<<(E01 + 255) / 256, 256, 0, stream>>>(inct, incs,      pt,  ps,  e01, rst, rss, E01);

    // ----------------------- normalized scatter ----------------------------
    edge_scatter<<<(E0  + 7) / 8, 256, 0, stream>>>(adj0, adj0 + E0, e0,  rs0, m0, out0, E0);  // x_0_to_0
    edge_scatter<<<(E1  + 7) / 8, 256, 0, stream>>>(adj1, adj1 + E1, e1,  rs1, m1, out1, E1);  // x_1_to_1
    edge_scatter<<<(E01 + 7) / 8, 256, 0, stream>>>(inct, incs,      e01, rst, sm, out0, E01); // x_1_to_0
    edge_scatter<<<(E01 + 7) / 8, 256, 0, stream>>>(incs, inct,      e01, rss, tm, out1, E01); // x_0_to_1
}